// MixedLogit_42949672960646
// MI455X (gfx1250) — compile-verified
//
#include <hip/hip_runtime.h>
#include <math.h>

typedef __attribute__((ext_vector_type(2))) float v2f;
typedef __attribute__((ext_vector_type(8))) float v8f;

#define NEGC (-1e30f)

__global__ __launch_bounds__(128) void MixedLogit_42949672960646_kernel(
    const float* __restrict__ X,      // [B,128,64]
    const int*   __restrict__ lens,   // [B]
    const float* __restrict__ thetas, // [64,64]
    const float* __restrict__ mw,     // [64]
    float* __restrict__ out)          // [B,128]
{
    __shared__ float s_theta[64 * 65];   // padded stride -> conflict-free fragment reads
    __shared__ float s_util[128 * 64];   // util[l][g]
    __shared__ float s_red[128];
    __shared__ float s_lse[64];
    __shared__ float s_w[64];

    const int tid  = threadIdx.x;
    const int b    = blockIdx.x;
    const int lane = tid & 31;
    const int wave = tid >> 5;
    const int len  = lens[b];

    // ---- Phase 0: stage thetas into LDS; mixture-weight log-softmax constant ----
    for (int idx = tid; idx < 64 * 64; idx += 128) {
        s_theta[(idx >> 6) * 65 + (idx & 63)] = thetas[idx];
    }
    if (tid == 0) {
        float m = mw[0];
        for (int g = 1; g < 64; ++g) m = fmaxf(m, mw[g]);
        float s = 0.f;
        for (int g = 0; g < 64; ++g) s += __expf(mw[g] - m);
        s_red[0] = m + __logf(s);   // logsumexp(mixture_weights)
    }
    __syncthreads();
    if (tid < 64) s_w[tid] = mw[tid] - s_red[0];

    // ---- Phase 1: util = X_b @ thetas^T via V_WMMA_F32_16X16X4_F32 ----
    // Wave 'wave' owns N-columns [n0, n0+16). M swept in 8 tiles of 16, K=64 in 16 steps of 4.
    const int n0   = wave * 16;
    const int lrow = lane & 15;
    const int lhi  = lane >> 4;     // 0 or 1: K-half / M-half selector per WMMA layouts

    // B fragments: B[k][n] = thetas[n][k]; VGPR j holds K = 4t + 2*lhi + j for row n0+lrow
    v2f bfrag[16];
    #pragma unroll
    for (int t = 0; t < 16; ++t) {
        const float* p = &s_theta[(n0 + lrow) * 65 + 4 * t + 2 * lhi];
        bfrag[t].x = p[0];
        bfrag[t].y = p[1];
    }

    const float* xb = X + (size_t)b * (128 * 64);
    for (int m = 0; m < 8; ++m) {
        const int m0 = m * 16;
        const float* arow = xb + (size_t)(m0 + lrow) * 64 + 2 * lhi;
        v8f acc0 = {};
        v8f acc1 = {};
        #pragma unroll
        for (int t = 0; t < 8; ++t) {
            const float2 f0 = *(const float2*)(arow + 4 * (2 * t));
            const float2 f1 = *(const float2*)(arow + 4 * (2 * t + 1));
            v2f a0, a1;
            a0.x = f0.x; a0.y = f0.y;
            a1.x = f1.x; a1.y = f1.y;
            // D = A*B + C, two independent accumulation chains over K
            acc0 = __builtin_amdgcn_wmma_f32_16x16x4_f32(false, a0, false, bfrag[2 * t],
                                                         (short)0, acc0, false, false);
            acc1 = __builtin_amdgcn_wmma_f32_16x16x4_f32(false, a1, false, bfrag[2 * t + 1],
                                                         (short)0, acc1, false, false);
        }
        // C/D layout: VGPR v, lanes 0-15 -> row m0+v, lanes 16-31 -> row m0+v+8; col n0+lrow
        #pragma unroll
        for (int v = 0; v < 8; ++v) {
            const int row = m0 + v + 8 * lhi;
            const float val = acc0[v] + acc1[v];
            s_util[row * 64 + n0 + lrow] = (row < len) ? val : NEGC;
        }
    }
    __syncthreads();

    // ---- Phase 2: per-component logsumexp over the L axis (two partials per column) ----
    {
        const int g    = tid & 63;
        const int r0   = (tid >> 6) * 64;
        float m = -3.402823466e38f;
        for (int r = 0; r < 64; ++r) m = fmaxf(m, s_util[(r0 + r) * 64 + g]);
        s_red[tid] = m;
        __syncthreads();
        if (tid < 64) s_lse[g] = fmaxf(s_red[g], s_red[g + 64]);
        __syncthreads();
        const float M = s_lse[g];
        float s = 0.f;
        for (int r = 0; r < 64; ++r) s += __expf(s_util[(r0 + r) * 64 + g] - M);
        s_red[tid] = s;
        __syncthreads();
        if (tid < 64) s_lse[g] = M + __logf(s_red[g] + s_red[g + 64]);
        __syncthreads();
    }

    // ---- Phase 3: per-row logsumexp over 64 components (wave32 butterfly) ----
    const float wlo = s_w[lane]      - s_lse[lane];
    const float whi = s_w[lane + 32] - s_lse[lane + 32];
    for (int i = 0; i < 32; ++i) {
        const int l = wave * 32 + i;
        const float a = s_util[l * 64 + lane]      + wlo;
        const float c = s_util[l * 64 + lane + 32] + whi;
        float m = fmaxf(a, c);
        #pragma unroll
        for (int off = 16; off > 0; off >>= 1) m = fmaxf(m, __shfl_xor(m, off, 32));
        float s = __expf(a - m) + __expf(c - m);
        #pragma unroll
        for (int off = 16; off > 0; off >>= 1) s += __shfl_xor(s, off, 32);
        if (lane == 0) {
            out[(size_t)b * 128 + l] = (l < len) ? (m + __logf(s)) : -INFINITY;
        }
    }
}

extern "C" void kernel_launch(void* const* d_in, const int* in_sizes, int n_in,
                              void* d_out, int out_size, void* d_ws, size_t ws_size,
                              hipStream_t stream) {
    (void)n_in; (void)out_size; (void)d_ws; (void)ws_size;
    const float* X      = (const float*)d_in[0];
    const int*   lens   = (const int*)d_in[1];
    const float* thetas = (const float*)d_in[2];
    const float* mw     = (const float*)d_in[3];
    float*       out    = (float*)d_out;
    const int B = in_sizes[1];  // 8192 batches (length array size)
    MixedLogit_42949672960646_kernel<<<dim3(B), dim3(128), 0, stream>>>(X, lens, thetas, mw, out);
}